// GRFormer_11364483465527
// MI455X (gfx1250) — compile-verified
//
#include <hip/hip_runtime.h>
#include <hip/hip_bf16.h>

typedef __attribute__((ext_vector_type(2))) float v2f;
typedef __attribute__((ext_vector_type(8))) float v8f;

#define WMMA_F32(acc, af, bf)                                                   \
  acc = __builtin_amdgcn_wmma_f32_16x16x4_f32(false, af, false, bf, (short)0,   \
                                              acc, false, false)

static constexpr int Bn = 4, Np = 2048, Cc = 256, Dd = 64, UPc = 4, KNNc = 20;
static constexpr float EPSc = 1e-5f;
static constexpr float BIGF = 3.402823466e38f;

// ---------------------------------------------------------------------------
// K1: value = Wv2 @ relu(Wv1 @ [key;query] + bv1) + bv2 + Wvs @ [key;query] + bvs
// One workgroup = 16 waves, one 16-column tile; hidden (256x16) staged in LDS.
// ---------------------------------------------------------------------------
__global__ __launch_bounds__(512) void k_mlpv(
    const float* __restrict__ key, const float* __restrict__ query,
    const float* __restrict__ wv1, const float* __restrict__ bv1,
    const float* __restrict__ wv2, const float* __restrict__ bv2,
    const float* __restrict__ wvs, const float* __restrict__ bvs,
    float* __restrict__ value) {
  __shared__ float h[256 * 16];
  const int b = blockIdx.y;
  const int n0 = blockIdx.x * 16;
  const int lane = threadIdx.x & 31;
  const int wid = threadIdx.x >> 5;  // 0..15
  const int m = lane & 15;
  const int kh = (lane >> 4) << 1;  // 0 or 2
  const int nc = lane & 15;
  const float* keyb = key + (size_t)b * Cc * Np;
  const float* qryb = query + (size_t)b * Cc * Np;
  const int mg = wid * 16 + m;
  const int mrb = wid * 16 + ((lane >> 4) << 3);

  {  // hidden rows [16w, 16w+16)
    v8f acc = {};
    for (int k0 = 0; k0 < 256; k0 += 4) {
      v2f af, bf;
      af.x = wv1[mg * 512 + k0 + kh];
      af.y = wv1[mg * 512 + k0 + kh + 1];
      bf.x = keyb[(k0 + kh) * Np + n0 + nc];
      bf.y = keyb[(k0 + kh + 1) * Np + n0 + nc];
      WMMA_F32(acc, af, bf);
    }
    for (int k0 = 0; k0 < 256; k0 += 4) {
      v2f af, bf;
      af.x = wv1[mg * 512 + 256 + k0 + kh];
      af.y = wv1[mg * 512 + 256 + k0 + kh + 1];
      bf.x = qryb[(k0 + kh) * Np + n0 + nc];
      bf.y = qryb[(k0 + kh + 1) * Np + n0 + nc];
      WMMA_F32(acc, af, bf);
    }
    __builtin_prefetch(wv2 + mg * 256, 0, 3);
#pragma unroll
    for (int r = 0; r < 8; ++r) {
      float x = acc[r] + bv1[mrb + r];
      h[(mrb + r) * 16 + nc] = x > 0.f ? x : 0.f;
    }
  }
  __syncthreads();
  {  // value rows [16w, 16w+16)
    v8f acc = {};
    for (int k0 = 0; k0 < 256; k0 += 4) {  // Wv2 @ h  (LDS)
      v2f af, bf;
      af.x = wv2[mg * 256 + k0 + kh];
      af.y = wv2[mg * 256 + k0 + kh + 1];
      bf.x = h[(k0 + kh) * 16 + nc];
      bf.y = h[(k0 + kh + 1) * 16 + nc];
      WMMA_F32(acc, af, bf);
    }
    for (int k0 = 0; k0 < 256; k0 += 4) {  // Wvs @ key-half
      v2f af, bf;
      af.x = wvs[mg * 512 + k0 + kh];
      af.y = wvs[mg * 512 + k0 + kh + 1];
      bf.x = keyb[(k0 + kh) * Np + n0 + nc];
      bf.y = keyb[(k0 + kh + 1) * Np + n0 + nc];
      WMMA_F32(acc, af, bf);
    }
    for (int k0 = 0; k0 < 256; k0 += 4) {  // Wvs @ query-half
      v2f af, bf;
      af.x = wvs[mg * 512 + 256 + k0 + kh];
      af.y = wvs[mg * 512 + 256 + k0 + kh + 1];
      bf.x = qryb[(k0 + kh) * Np + n0 + nc];
      bf.y = qryb[(k0 + kh + 1) * Np + n0 + nc];
      WMMA_F32(acc, af, bf);
    }
    float* vout = value + (size_t)b * Cc * Np;
#pragma unroll
    for (int r = 0; r < 8; ++r)
      vout[(mrb + r) * Np + n0 + nc] = acc[r] + bv2[mrb + r] + bvs[mrb + r];
  }
}

// ---------------------------------------------------------------------------
// K2: k_f / q_f / v_f projections: (64x256) @ (256x2048) per batch.
// ---------------------------------------------------------------------------
__global__ __launch_bounds__(128) void k_proj(
    const float* __restrict__ key, const float* __restrict__ query,
    const float* __restrict__ value, const float* __restrict__ wk,
    const float* __restrict__ bk, const float* __restrict__ wq,
    const float* __restrict__ bq, const float* __restrict__ wval,
    const float* __restrict__ bval, float* __restrict__ kf,
    float* __restrict__ qf, float* __restrict__ vf) {
  const int b = blockIdx.z, p = blockIdx.y;
  const int n0 = blockIdx.x * 16;
  const float *W, *bia, *X;
  float* Y;
  if (p == 0) { W = wk;   bia = bk;   X = key;   Y = kf; }
  else if (p == 1) { W = wq;   bia = bq;   X = query; Y = qf; }
  else { W = wval; bia = bval; X = value; Y = vf; }
  X += (size_t)b * Cc * Np;
  Y += (size_t)b * Dd * Np;
  const int lane = threadIdx.x & 31, wid = threadIdx.x >> 5;
  const int m = lane & 15, kh = (lane >> 4) << 1, nc = lane & 15;
  const int mg = wid * 16 + m;
  v8f acc = {};
  for (int k0 = 0; k0 < 256; k0 += 4) {
    v2f af, bf;
    af.x = W[mg * 256 + k0 + kh];
    af.y = W[mg * 256 + k0 + kh + 1];
    bf.x = X[(k0 + kh) * Np + n0 + nc];
    bf.y = X[(k0 + kh + 1) * Np + n0 + nc];
    WMMA_F32(acc, af, bf);
  }
  const int mrb = wid * 16 + ((lane >> 4) << 3);
#pragma unroll
  for (int r = 0; r < 8; ++r)
    Y[(mrb + r) * Np + n0 + nc] = acc[r] + bia[mrb + r];
}

// ---------------------------------------------------------------------------
// K3: exact KNN (20 smallest sq distances, include self). One WG per point.
// ---------------------------------------------------------------------------
__global__ __launch_bounds__(256) void k_knn(const float* __restrict__ pos,
                                             int* __restrict__ idx) {
  __shared__ float dist[2048];
  __shared__ float rv[256];
  __shared__ int ri[256];
  const int b = blockIdx.y, n = blockIdx.x, t = threadIdx.x;
  const float* pb = pos + (size_t)b * 3 * Np;
  const float px = pb[n], py = pb[Np + n], pz = pb[2 * Np + n];
  for (int j = t; j < Np; j += 256) {
    float dx = px - pb[j], dy = py - pb[Np + j], dz = pz - pb[2 * Np + j];
    dist[j] = dx * dx + dy * dy + dz * dz;
  }
  __syncthreads();
  for (int s = 0; s < KNNc; ++s) {
    float bv = BIGF;
    int bi = 0x7fffffff;
    for (int j = t; j < Np; j += 256) {
      float d = dist[j];
      if (d < bv || (d == bv && j < bi)) { bv = d; bi = j; }
    }
    rv[t] = bv;
    ri[t] = bi;
    __syncthreads();
    for (int off = 128; off > 0; off >>= 1) {
      if (t < off) {
        float ov = rv[t + off];
        int oi = ri[t + off];
        if (ov < rv[t] || (ov == rv[t] && oi < ri[t])) { rv[t] = ov; ri[t] = oi; }
      }
      __syncthreads();
    }
    if (t == 0) {
      idx[((size_t)b * Np + n) * KNNc + s] = ri[0];
      dist[ri[0]] = BIGF;
    }
    __syncthreads();
  }
}

// ---------------------------------------------------------------------------
// K4: fused attention per (b, n): pos_mlp -> qk_rel -> attn_mlp -> ConvT ->
// softmax -> aggregation. 8 waves. Everything lives in 62 KB of LDS.
// ---------------------------------------------------------------------------
__global__ __launch_bounds__(256) void k_attn(
    const float* __restrict__ pos, const int* __restrict__ idx,
    const float* __restrict__ kf, const float* __restrict__ qf,
    const float* __restrict__ vf, const float* __restrict__ pw1,
    const float* __restrict__ pb1, const float* __restrict__ pg,
    const float* __restrict__ pbt, const float* __restrict__ prm,
    const float* __restrict__ prv, const float* __restrict__ pw2,
    const float* __restrict__ pb2, const float* __restrict__ aw1,
    const float* __restrict__ ab1, const float* __restrict__ ag,
    const float* __restrict__ abt, const float* __restrict__ arm,
    const float* __restrict__ arv, const float* __restrict__ ctw,
    const float* __restrict__ ctb, float* __restrict__ agg) {
  __shared__ float regA[256 * 32];   // h1 (64x32) -> a (256x32)
  __shared__ float regCT[256 * 20];  // s (64x32)  -> ct/attn (256x20)
  __shared__ float regPE[64 * 20];   // pos_emb
  __shared__ float regV[64 * 20];    // v_g
  __shared__ int nidx[KNNc];
  const int b = blockIdx.y, n = blockIdx.x, t = threadIdx.x;
  const int lane = t & 31, wid = t >> 5;
  const int mloc = lane & 15, kh = (lane >> 4) << 1, nc = lane & 15;
  const float* posb = pos + (size_t)b * 3 * Np;
  const float* kfb = kf + (size_t)b * Dd * Np;
  const float* qfb = qf + (size_t)b * Dd * Np;
  const float* vfb = vf + (size_t)b * Dd * Np;

  if (t < KNNc) nidx[t] = idx[((size_t)b * Np + n) * KNNc + t];
  __syncthreads();

  // phase 1: h1 = relu(bn(pw1 @ pos_rel + pb1))   (64 x 32, cols>=20 zero)
  for (int i = t; i < 64 * 32; i += 256) {
    int c = i >> 5, k = i & 31;
    float val = 0.f;
    if (k < KNNc) {
      int j = nidx[k];
      float acc = 0.f;
#pragma unroll
      for (int d = 0; d < 3; ++d)
        acc += pw1[c * 3 + d] * (posb[d * Np + n] - posb[d * Np + j]);
      float sc = pg[c] * rsqrtf(prv[c] + EPSc);
      val = sc * (acc + pb1[c] - prm[c]) + pbt[c];
      val = val > 0.f ? val : 0.f;
    }
    regA[i] = val;
  }
  __syncthreads();

  // phase 2: pos_emb = pw2 @ h1 + pb2   (WMMA, 8 tiles / 8 waves)
  {
    const int rt = wid & 3, ctile = wid >> 2;
    const int mg = rt * 16 + mloc;
    v8f acc = {};
    for (int k0 = 0; k0 < 64; k0 += 4) {
      v2f af, bf;
      af.x = pw2[mg * 64 + k0 + kh];
      af.y = pw2[mg * 64 + k0 + kh + 1];
      bf.x = regA[(k0 + kh) * 32 + ctile * 16 + nc];
      bf.y = regA[(k0 + kh + 1) * 32 + ctile * 16 + nc];
      WMMA_F32(acc, af, bf);
    }
    const int mrb = rt * 16 + ((lane >> 4) << 3);
    const int col = ctile * 16 + nc;
    if (col < KNNc) {
#pragma unroll
      for (int r = 0; r < 8; ++r)
        regPE[(mrb + r) * KNNc + col] = acc[r] + pb2[mrb + r];
    }
  }
  __syncthreads();

  // phase 3: s = q_f - k_g + pos_emb   (64 x 32, cols>=20 zero)
  for (int i = t; i < 64 * 32; i += 256) {
    int c = i >> 5, k = i & 31;
    float val = 0.f;
    if (k < KNNc)
      val = qfb[c * Np + n] - kfb[c * Np + nidx[k]] + regPE[c * KNNc + k];
    regCT[i] = val;
  }
  __syncthreads();

  // phase 4: a = relu(bn(aw1 @ s + ab1))   (256 x 32; 32 tiles, 4 per wave)
  for (int i2 = 0; i2 < 4; ++i2) {
    const int tile = wid + 8 * i2;
    const int rt = tile & 15, ctile = tile >> 4;
    const int mg = rt * 16 + mloc;
    v8f acc = {};
    for (int k0 = 0; k0 < 64; k0 += 4) {
      v2f af, bf;
      af.x = aw1[mg * 64 + k0 + kh];
      af.y = aw1[mg * 64 + k0 + kh + 1];
      bf.x = regCT[(k0 + kh) * 32 + ctile * 16 + nc];
      bf.y = regCT[(k0 + kh + 1) * 32 + ctile * 16 + nc];
      WMMA_F32(acc, af, bf);
    }
    const int mrb = rt * 16 + ((lane >> 4) << 3);
    const int col = ctile * 16 + nc;
#pragma unroll
    for (int r = 0; r < 8; ++r) {
      int mm = mrb + r;
      float sc = ag[mm] * rsqrtf(arv[mm] + EPSc);
      float val = sc * (acc[r] + ab1[mm] - arm[mm]) + abt[mm];
      regA[mm * 32 + col] = val > 0.f ? val : 0.f;
    }
  }
  __syncthreads();

  // phase 5: ct[(o*4+r), k] = sum_c ctw[c,o,r]*a[c,k] + ctb[o]
  for (int i2 = 0; i2 < 4; ++i2) {
    const int tile = wid + 8 * i2;
    const int rt = tile & 15, ctile = tile >> 4;
    const int mg = rt * 16 + mloc;
    v8f acc = {};
    for (int k0 = 0; k0 < 256; k0 += 4) {
      v2f af, bf;
      af.x = ctw[(k0 + kh) * 256 + mg];       // transposed ctw access
      af.y = ctw[(k0 + kh + 1) * 256 + mg];
      bf.x = regA[(k0 + kh) * 32 + ctile * 16 + nc];
      bf.y = regA[(k0 + kh + 1) * 32 + ctile * 16 + nc];
      WMMA_F32(acc, af, bf);
    }
    const int mrb = rt * 16 + ((lane >> 4) << 3);
    const int col = ctile * 16 + nc;
    if (col < KNNc) {
#pragma unroll
      for (int r = 0; r < 8; ++r) {
        int mm = mrb + r;
        regCT[mm * KNNc + col] = acc[r] + ctb[mm >> 2];
      }
    }
  }
  __syncthreads();

  // phase 6: softmax over neighbors; row m = t
  {
    float vals[KNNc];
    float mx = -BIGF;
#pragma unroll
    for (int k = 0; k < KNNc; ++k) {
      vals[k] = regCT[t * KNNc + k];
      mx = fmaxf(mx, vals[k]);
    }
    float sum = 0.f;
#pragma unroll
    for (int k = 0; k < KNNc; ++k) {
      vals[k] = __expf(vals[k] - mx);
      sum += vals[k];
    }
    float inv = 1.f / sum;
#pragma unroll
    for (int k = 0; k < KNNc; ++k) regCT[t * KNNc + k] = vals[k] * inv;
  }
  // phase 7: v_g = gather(v_f) + pos_emb
  for (int i = t; i < 64 * KNNc; i += 256) {
    int o = i / KNNc, k = i % KNNc;
    regV[i] = vfb[o * Np + nidx[k]] + regPE[i];
  }
  __syncthreads();
  // phase 8: agg[o, n*4 + r] = sum_k attn * v
  {
    const int o = t >> 2, r = t & 3;
    float s = 0.f;
#pragma unroll
    for (int k = 0; k < KNNc; ++k) s += regCT[t * KNNc + k] * regV[o * KNNc + k];
    agg[(size_t)b * Dd * (Np * UPc) + (size_t)o * (Np * UPc) + n * UPc + r] = s;
  }
}

// ---------------------------------------------------------------------------
// K5: y = we @ agg + be + repeat(value, UP)   (256x64 @ 64x8192 per batch)
// ---------------------------------------------------------------------------
__global__ __launch_bounds__(512) void k_final(
    const float* __restrict__ agg, const float* __restrict__ we,
    const float* __restrict__ be, const float* __restrict__ value,
    float* __restrict__ out) {
  const int b = blockIdx.y, n0 = blockIdx.x * 16;
  const int NU = Np * UPc;
  const int lane = threadIdx.x & 31, wid = threadIdx.x >> 5;
  const int m = lane & 15, kh = (lane >> 4) << 1, nc = lane & 15;
  const float* Bm = agg + (size_t)b * Dd * NU;
  const int mg = wid * 16 + m;
  v8f acc = {};
  for (int k0 = 0; k0 < 64; k0 += 4) {
    v2f af, bf;
    af.x = we[mg * 64 + k0 + kh];
    af.y = we[mg * 64 + k0 + kh + 1];
    bf.x = Bm[(k0 + kh) * NU + n0 + nc];
    bf.y = Bm[(k0 + kh + 1) * NU + n0 + nc];
    WMMA_F32(acc, af, bf);
  }
  const int mrb = wid * 16 + ((lane >> 4) << 3);
  const float* valb = value + (size_t)b * Cc * Np;
  float* ob = out + (size_t)b * Cc * NU;
  const int col = n0 + nc;
#pragma unroll
  for (int r = 0; r < 8; ++r) {
    int mm = mrb + r;
    ob[(size_t)mm * NU + col] = acc[r] + be[mm] + valb[mm * Np + (col >> 2)];
  }
}

// ---------------------------------------------------------------------------
extern "C" void kernel_launch(void* const* d_in, const int* in_sizes, int n_in,
                              void* d_out, int out_size, void* d_ws,
                              size_t ws_size, hipStream_t stream) {
  const float* pos = (const float*)d_in[0];
  const float* key = (const float*)d_in[1];
  const float* query = (const float*)d_in[2];
  const float* wv1 = (const float*)d_in[3];
  const float* bv1 = (const float*)d_in[4];
  const float* wv2 = (const float*)d_in[5];
  const float* bv2 = (const float*)d_in[6];
  const float* wvs = (const float*)d_in[7];
  const float* bvs = (const float*)d_in[8];
  const float* wk = (const float*)d_in[9];
  const float* bk = (const float*)d_in[10];
  const float* wq = (const float*)d_in[11];
  const float* bq = (const float*)d_in[12];
  const float* wval = (const float*)d_in[13];
  const float* bval = (const float*)d_in[14];
  const float* pw1 = (const float*)d_in[15];
  const float* pb1 = (const float*)d_in[16];
  const float* pg = (const float*)d_in[17];
  const float* pbt = (const float*)d_in[18];
  const float* prm = (const float*)d_in[19];
  const float* prv = (const float*)d_in[20];
  const float* pw2 = (const float*)d_in[21];
  const float* pb2 = (const float*)d_in[22];
  const float* aw1 = (const float*)d_in[23];
  const float* ab1 = (const float*)d_in[24];
  const float* ag = (const float*)d_in[25];
  const float* abt = (const float*)d_in[26];
  const float* arm = (const float*)d_in[27];
  const float* arv = (const float*)d_in[28];
  const float* ctw = (const float*)d_in[29];
  const float* ctb = (const float*)d_in[30];
  const float* we = (const float*)d_in[31];
  const float* be = (const float*)d_in[32];

  float* ws = (float*)d_ws;
  float* value = ws;                       // 4*256*2048  (8 MB)
  float* kf = value + 4 * 256 * 2048;      // 4*64*2048   (2 MB)
  float* qf = kf + 4 * 64 * 2048;
  float* vf = qf + 4 * 64 * 2048;
  float* agg = vf + 4 * 64 * 2048;         // 4*64*8192   (8 MB)
  int* idx = (int*)(agg + 4 * 64 * 8192);  // 4*2048*20 ints

  k_mlpv<<<dim3(128, 4), 512, 0, stream>>>(key, query, wv1, bv1, wv2, bv2, wvs,
                                           bvs, value);
  k_knn<<<dim3(2048, 4), 256, 0, stream>>>(pos, idx);
  k_proj<<<dim3(128, 3, 4), 128, 0, stream>>>(key, query, value, wk, bk, wq, bq,
                                              wval, bval, kf, qf, vf);
  k_attn<<<dim3(2048, 4), 256, 0, stream>>>(pos, idx, kf, qf, vf, pw1, pb1, pg,
                                            pbt, prm, prv, pw2, pb2, aw1, ab1,
                                            ag, abt, arm, arv, ctw, ctb, agg);
  k_final<<<dim3(512, 4), 512, 0, stream>>>(agg, we, be, value, (float*)d_out);
}